// TAGGraphReg_13675175870529
// MI455X (gfx1250) — compile-verified
//
#include <hip/hip_runtime.h>

typedef __attribute__((ext_vector_type(2))) float v2f;
typedef __attribute__((ext_vector_type(4))) float v4f;
typedef __attribute__((ext_vector_type(8))) float v8f;

#define N_NODES 100000
#define N_EDGES 3200000
#define G_GRAPHS 64
#define HD 16
#define K_HOPS 4
#define LRELU_SLOPE 0.01f
#define NTILES (N_NODES / 16)   // 6250, exact
#define CDIV(a, b) (((a) + (b) - 1) / (b))

__global__ void zero_f32(float* __restrict__ p, int n) {
  int i = blockIdx.x * blockDim.x + threadIdx.x;
  if (i < n) p[i] = 0.0f;
}

__global__ void deg_kernel(const int* __restrict__ dst, const float* __restrict__ ea,
                           float* __restrict__ deg) {
  int e = blockIdx.x * blockDim.x + threadIdx.x;
  if (e >= N_EDGES) return;
  atomicAdd(&deg[dst[e]], ea[e * 7 + 6]);
}

__global__ void dinv_kernel(float* __restrict__ d, int n) {
  int i = blockIdx.x * blockDim.x + threadIdx.x;
  if (i >= n) return;
  float v = d[i];
  d[i] = (v > 0.0f) ? rsqrtf(v) : 0.0f;
}

__global__ void norm_kernel(const int* __restrict__ src, const int* __restrict__ dst,
                            const float* __restrict__ ea, const float* __restrict__ dinv,
                            float* __restrict__ nrm) {
  int e = blockIdx.x * blockDim.x + threadIdx.x;
  if (e >= N_EDGES) return;
  nrm[e] = dinv[src[e]] * ea[e * 7 + 6] * dinv[dst[e]];
}

template <int HF>
__global__ void propagate_kernel(const float* __restrict__ hin, const int* __restrict__ src,
                                 const int* __restrict__ dst, const float* __restrict__ nrm,
                                 float* __restrict__ hout) {
  int e = blockIdx.x * blockDim.x + threadIdx.x;
  if (e >= N_EDGES) return;
  int s = src[e], d = dst[e];
  float nv = nrm[e];
  if constexpr (HF == 2) {
    v2f r = *(const v2f*)(hin + s * 2);
    atomicAdd(&hout[d * 2 + 0], r.x * nv);
    atomicAdd(&hout[d * 2 + 1], r.y * nv);
  } else {
#pragma unroll
    for (int q = 0; q < 4; ++q) {
      v4f r = *(const v4f*)(hin + s * 16 + q * 4);   // global_load_b128
#pragma unroll
      for (int j = 0; j < 4; ++j)
        atomicAdd(&hout[d * 16 + q * 4 + j], r[j] * nv);
    }
  }
}

// yp (tile-packed scratch) accumulation of h_tile[16][Kin] @ W[Kin][16] via
// V_WMMA_F32_16X16X4_F32. One wave per 16-node tile, EXEC all-ones.
// f32 operand layouts per ISA §7.12.2:
//   A 16x4: lanes 0-15 hold K=0,1 (vgpr0,1); lanes 16-31 hold K=2,3
//   B 4x16: vgpr0 = rows K=0 (lanes 0-15) / K=2 (lanes 16-31); vgpr1 = K=1 / K=3
//   C/D 16x16: vgpr r = row r+half*8, col = lane%16
// Packed layout: yp[tile*256 + lane*8 + r]  (each lane's 8 C elems contiguous
// -> 2x b128 load + 2x b128 store instead of 16 scattered b32).
// ACC=false: C starts at zero (k=0, saves zero-pass). FIN=true: apply bias +
// LeakyReLU and emit node-major layer output (fuses bias_leaky kernel).
template <int Kin, bool ACC, bool FIN>
__global__ __launch_bounds__(256) void
tag_matmul_kernel(const float* __restrict__ h, const float* __restrict__ W,
                  float* __restrict__ yp, const float* __restrict__ bias,
                  float* __restrict__ outh) {
  constexpr int KP = (Kin + 3) & ~3;          // K padded to multiple of 4
  constexpr int CH = KP / 4;                  // # of WMMA chunks (1 or 4)
  // Pair-major staging: Wp[p*16+n] = {W[2p][n], W[2p+1][n]}, zero-padded
  __shared__ v2f Wp[(KP / 2) * 16];
  for (int i = threadIdx.x; i < (KP / 2) * 16; i += blockDim.x) {
    int p = i >> 4, n = i & 15;
    v2f w;
    w.x = (2 * p + 0 < Kin) ? W[(2 * p + 0) * 16 + n] : 0.0f;
    w.y = (2 * p + 1 < Kin) ? W[(2 * p + 1) * 16 + n] : 0.0f;
    Wp[i] = w;
  }
  __syncthreads();

  int wave = (int)(blockIdx.x * blockDim.x + threadIdx.x) >> 5;
  int lane = threadIdx.x & 31;
  if (wave >= NTILES) return;                 // wave-uniform guard, EXEC all-ones
  int half = lane >> 4;                       // 0 = lanes 0-15, 1 = lanes 16-31
  int col  = lane & 15;                       // N index (B/C/D), M index (A)
  int node = wave * 16 + col;

  float* ytile = yp + wave * 256 + lane * 8;

  v8f c;
  if constexpr (ACC) {
    v4f c0 = *(const v4f*)(ytile);            // global_load_b128
    v4f c1 = *(const v4f*)(ytile + 4);
#pragma unroll
    for (int r = 0; r < 4; ++r) { c[r] = c0[r]; c[r + 4] = c1[r]; }
  } else {
    c = (v8f){};
  }

#pragma unroll
  for (int ch = 0; ch < CH; ++ch) {
    v2f a;
    if constexpr (Kin == 2) {
      // Only K=0,1 exist; upper-half lanes (K=2,3) contribute zero.
      v2f av = *(const v2f*)(h + node * 2);   // in-bounds for all lanes
      a.x = half ? 0.0f : av.x;               // v_cndmask, no exec games
      a.y = half ? 0.0f : av.y;
    } else {
      int f0 = ch * 4 + half * 2;             // even offset -> aligned b64 load
      a = *(const v2f*)(h + node * Kin + f0);
    }
    v2f b = Wp[(ch * 2 + half) * 16 + col];   // ds_load_b64
    // 8-arg form: (neg_a, A, neg_b, B, c_mod, C, reuse_a, reuse_b)
    c = __builtin_amdgcn_wmma_f32_16x16x4_f32(false, a, false, b, (short)0, c,
                                              false, false);
  }

  if constexpr (FIN) {
    float bb = bias[col];
#pragma unroll
    for (int r = 0; r < 8; ++r) {
      float v = c[r] + bb;
      v = (v >= 0.0f) ? v : LRELU_SLOPE * v;
      outh[(wave * 16 + r + half * 8) * HD + col] = v;  // node-major output
    }
  } else {
    v4f c0, c1;
#pragma unroll
    for (int r = 0; r < 4; ++r) { c0[r] = c[r]; c1[r] = c[r + 4]; }
    *(v4f*)(ytile) = c0;                      // global_store_b128
    *(v4f*)(ytile + 4) = c1;
  }
}

__global__ void pool_kernel(const float* __restrict__ h, const int* __restrict__ batch,
                            float* __restrict__ pooled, float* __restrict__ counts) {
  int i = blockIdx.x * blockDim.x + threadIdx.x;
  if (i >= N_NODES) return;
  int g = batch[i];
  atomicAdd(&counts[g], 1.0f);
#pragma unroll
  for (int q = 0; q < 4; ++q) {
    v4f r = *(const v4f*)(h + i * 16 + q * 4);
#pragma unroll
    for (int j = 0; j < 4; ++j)
      atomicAdd(&pooled[g * HD + q * 4 + j], r[j]);
  }
}

__global__ void readout_kernel(const float* __restrict__ pooled, const float* __restrict__ counts,
                               const float* __restrict__ Ws, const float* __restrict__ bs,
                               float* __restrict__ out) {
  int g = threadIdx.x;
  if (g >= G_GRAPHS) return;
  float inv = 1.0f / fmaxf(counts[g], 1.0f);
  float acc = 0.0f;
#pragma unroll
  for (int f = 0; f < HD; ++f) acc += pooled[g * HD + f] * inv * Ws[f];
  out[g] = acc + bs[0];
}

extern "C" void kernel_launch(void* const* d_in, const int* in_sizes, int n_in,
                              void* d_out, int out_size, void* d_ws, size_t ws_size,
                              hipStream_t stream) {
  const float* x     = (const float*)d_in[0];          // N x 2
  const int*   ei    = (const int*)d_in[1];            // 2 x E
  const float* ea    = (const float*)d_in[2];          // E x 7
  const int*   batch = (const int*)d_in[3];            // N
  const float* W1    = (const float*)d_in[4];          // 5 x 2 x 16
  const float* b1    = (const float*)d_in[5];          // 16
  const float* Wl    = (const float*)d_in[6];          // 2 x 5 x 16 x 16
  const float* bl    = (const float*)d_in[7];          // 2 x 16
  const float* Ws    = (const float*)d_in[8];          // 16 x 1
  const float* bs    = (const float*)d_in[9];          // 1
  float* out = (float*)d_out;

  const int* srcI = ei;
  const int* dstI = ei + N_EDGES;

  // Workspace layout (floats)
  float* wsf    = (float*)d_ws;
  float* nrm    = wsf;                         // E
  float* dinv   = nrm + N_EDGES;               // N  (deg, then dinv in place)
  float* hA     = dinv + N_NODES;              // N*16 (hop ping)
  float* hB     = hA + N_NODES * HD;           // N*16 (hop pong)
  float* yp     = hB + N_NODES * HD;           // N*16 (tile-packed accumulator)
  float* hIO    = yp + N_NODES * HD;           // N*16 (layer output / next input)
  float* pooled = hIO + N_NODES * HD;          // 64*16
  float* counts = pooled + G_GRAPHS * HD;      // 64

  const int TB = 256;
  const int EB = CDIV(N_EDGES, TB);
  const int NB = CDIV(N_NODES, TB);
  const int MMB = CDIV(NTILES * 32, TB);

  // ---- normalization coefficients ----
  zero_f32<<<NB, TB, 0, stream>>>(dinv, N_NODES);
  deg_kernel<<<EB, TB, 0, stream>>>(dstI, ea, dinv);
  dinv_kernel<<<NB, TB, 0, stream>>>(dinv, N_NODES);
  norm_kernel<<<EB, TB, 0, stream>>>(srcI, dstI, ea, dinv, nrm);

  // ---- one TAG layer ----
  auto run_layer = [&](const float* in, int Kin, const float* W, const float* b,
                       float* outh) {
    const int wstride = Kin * HD;
    float* bufs[2] = {hA, hB};
    if (Kin == 2) {
      tag_matmul_kernel<2, false, false><<<MMB, TB, 0, stream>>>(in, W, yp, b, outh);
      const float* p = in;
      for (int k = 1; k <= K_HOPS; ++k) {
        float* nb = bufs[(k - 1) & 1];
        zero_f32<<<CDIV(N_NODES * 2, TB), TB, 0, stream>>>(nb, N_NODES * 2);
        propagate_kernel<2><<<EB, TB, 0, stream>>>(p, srcI, dstI, nrm, nb);
        if (k < K_HOPS)
          tag_matmul_kernel<2, true, false><<<MMB, TB, 0, stream>>>(
              nb, W + k * wstride, yp, b, outh);
        else
          tag_matmul_kernel<2, true, true><<<MMB, TB, 0, stream>>>(
              nb, W + k * wstride, yp, b, outh);
        p = nb;
      }
    } else {
      tag_matmul_kernel<16, false, false><<<MMB, TB, 0, stream>>>(in, W, yp, b, outh);
      const float* p = in;
      for (int k = 1; k <= K_HOPS; ++k) {
        float* nb = bufs[(k - 1) & 1];
        zero_f32<<<CDIV(N_NODES * HD, TB), TB, 0, stream>>>(nb, N_NODES * HD);
        propagate_kernel<16><<<EB, TB, 0, stream>>>(p, srcI, dstI, nrm, nb);
        if (k < K_HOPS)
          tag_matmul_kernel<16, true, false><<<MMB, TB, 0, stream>>>(
              nb, W + k * wstride, yp, b, outh);
        else
          tag_matmul_kernel<16, true, true><<<MMB, TB, 0, stream>>>(
              nb, W + k * wstride, yp, b, outh);
        p = nb;
      }
    }
  };

  run_layer(x, 2, W1, b1, hIO);                            // layer 1: F=2 -> 16
  run_layer(hIO, HD, Wl + 0 * 5 * HD * HD, bl + 0, hIO);   // layer 2
  run_layer(hIO, HD, Wl + 1 * 5 * HD * HD, bl + HD, hIO);  // layer 3

  // ---- mean pool per graph + linear readout ----
  zero_f32<<<1, TB, 0, stream>>>(pooled, G_GRAPHS * HD + G_GRAPHS);
  pool_kernel<<<NB, TB, 0, stream>>>(hIO, batch, pooled, counts);
  readout_kernel<<<1, 64, 0, stream>>>(pooled, counts, Ws, bs, out);
  (void)in_sizes; (void)n_in; (void)out_size; (void)ws_size;
}